// HybridLoss3_58076547776957
// MI455X (gfx1250) — compile-verified
//
#include <hip/hip_runtime.h>
#include <hip/hip_bf16.h>
#include <math.h>

// ---------------------------------------------------------------------------
// Problem constants (match reference)
// ---------------------------------------------------------------------------
#define NUM_CLASSES 32768
#define FEAT_DIM    256
#define BATCH_N     2048
#define MARGIN_V    0.3f
#define CONF_V      0.9f
#define SMOOTH_V    0.1f
#define CENTER_W    5e-4f

typedef __attribute__((ext_vector_type(2))) float v2f;
typedef __attribute__((ext_vector_type(4))) float v4f;
typedef __attribute__((ext_vector_type(8))) float v8f;

// ---------------------------------------------------------------------------
// Kernel 0: zero accumulators + per-row ||e||^2 (one wave32 per row)
// grid = 256 blocks x 256 threads (8 rows / block)
// ---------------------------------------------------------------------------
__global__ __launch_bounds__(256) void prep_kernel(const float* __restrict__ E,
                                                   float* __restrict__ sq,
                                                   float* __restrict__ acc) {
  if (blockIdx.x == 0 && threadIdx.x < 8) acc[threadIdx.x] = 0.0f;
  const int wave = threadIdx.x >> 5;
  const int lane = threadIdx.x & 31;
  const int row  = blockIdx.x * 8 + wave;
  const float* __restrict__ x = E + (size_t)row * FEAT_DIM;
  float s = 0.0f;
#pragma unroll
  for (int k = lane; k < FEAT_DIM; k += 32) { float v = x[k]; s += v * v; }
#pragma unroll
  for (int off = 16; off > 0; off >>= 1) s += __shfl_xor(s, off, 32);
  if (lane == 0) sq[row] = s;
}

// ---------------------------------------------------------------------------
// Kernel 1: label-smoothing loss. One block (256 thr) per row.
// Single streaming pass: online logsumexp + sum + target-logit capture.
// 256 MB read-once (> 192 MB L2) -> non-temporal b128 loads so the stream
// doesn't cycle L2 (keeps E/centers resident for the later kernels).
// ---------------------------------------------------------------------------
__global__ __launch_bounds__(256) void smooth_kernel(const float* __restrict__ logits,
                                                     const int* __restrict__ tgt,
                                                     float* __restrict__ acc) {
  __shared__ float sm[256], ss[256], sx[256];
  __shared__ float xts;
  const int row = blockIdx.x;
  const int tid = threadIdx.x;
  const int t   = tgt[row];
  const v4f* __restrict__ p = (const v4f*)(logits + (size_t)row * NUM_CLASSES);

  float m = -INFINITY, s = 0.0f, sum = 0.0f, xt = 0.0f;
  bool has_t = false;
#pragma unroll 4
  for (int it = 0; it < NUM_CLASSES / (4 * 256); ++it) {
    const int i4 = tid + 256 * it;
    v4f x = __builtin_nontemporal_load(&p[i4]);   // global_load_b128, TH=NT
#pragma unroll
    for (int c = 0; c < 4; ++c) {
      float xk = x[c];
      sum += xk;
      if (xk > m) { s = s * __expf(m - xk) + 1.0f; m = xk; }
      else        { s += __expf(xk - m); }
      if (i4 * 4 + c == t) { xt = xk; has_t = true; }
    }
  }
  if (has_t) xts = xt;           // exactly one writer per block
  sm[tid] = m; ss[tid] = s; sx[tid] = sum;
  __syncthreads();
  for (int off = 128; off > 0; off >>= 1) {
    if (tid < off) {
      float m2 = sm[tid + off], s2 = ss[tid + off];
      float M  = fmaxf(sm[tid], m2);
      ss[tid]  = ss[tid] * __expf(sm[tid] - M) + s2 * __expf(m2 - M);
      sm[tid]  = M;
      sx[tid] += sx[tid + off];
    }
    __syncthreads();
  }
  if (tid == 0) {
    float logZ = sm[0] + __logf(ss[0]);
    float poly = logZ - CONF_V * xts - SMOOTH_V * (sx[0] * (1.0f / NUM_CLASSES));
    atomicAdd(&acc[0], poly);
  }
}

// ---------------------------------------------------------------------------
// Kernel 2: triplet loss via f32 WMMA Gram matrix.
// grid = 128 blocks (16-row i-tile each) x 256 threads (8 waves).
// A tile (16x256) staged in LDS; each wave sweeps disjoint 16-col j-tiles,
// accumulating G = E E^T with V_WMMA_F32_16X16X4_F32, K in steps of 4.
// C/D layout: lane l, slot v -> (M = v + 8*(l>>4), N = l&15).
// Distances only feed max/min/hinge -> raw v_sqrt_f32 suffices.
// ---------------------------------------------------------------------------
__global__ __launch_bounds__(256) void triplet_kernel(const float* __restrict__ E,
                                                      const int* __restrict__ tgt,
                                                      const float* __restrict__ sq,
                                                      float* __restrict__ acc) {
  __shared__ float At[16 * FEAT_DIM];
  __shared__ float sqA[16];
  __shared__ int   tgA[16];
  __shared__ float apL[8][16];
  __shared__ float anL[8][16];

  const int tid = threadIdx.x;
  const int i0  = blockIdx.x * 16;

  for (int idx = tid; idx < 16 * FEAT_DIM; idx += 256)
    At[idx] = E[(size_t)i0 * FEAT_DIM + idx];
  if (tid < 16) { sqA[tid] = sq[i0 + tid]; tgA[tid] = tgt[i0 + tid]; }
  __syncthreads();

  const int wave = tid >> 5;
  const int lane = tid & 31;
  const int hh   = lane >> 4;   // half-wave select
  const int lr   = lane & 15;   // N within tile

  float ap[8], an[8];
#pragma unroll
  for (int v = 0; v < 8; ++v) { ap[v] = -INFINITY; an[v] = INFINITY; }

  for (int jt = wave; jt < BATCH_N / 16; jt += 8) {
    const int j0 = jt * 16;
    const int n  = j0 + lr;
    const float* __restrict__ Bp = E + (size_t)n * FEAT_DIM;   // B col n = E row n
    v8f c = {0.f, 0.f, 0.f, 0.f, 0.f, 0.f, 0.f, 0.f};
#pragma unroll 8
    for (int kb = 0; kb < FEAT_DIM; kb += 4) {
      const int ka = kb + (hh << 1);       // A/B 16x4 f32 layout: v0:{K=0|K=2}, v1:{K=1|K=3}
      v2f a, b;
      a.x = At[lr * FEAT_DIM + ka];
      a.y = At[lr * FEAT_DIM + ka + 1];
      b.x = Bp[ka];
      b.y = Bp[ka + 1];
      c = __builtin_amdgcn_wmma_f32_16x16x4_f32(false, a, false, b, (short)0, c,
                                                false, false);
    }
    const float sqj = sq[n];
    const int   tj  = tgt[n];
#pragma unroll
    for (int v = 0; v < 8; ++v) {
      const int M  = v + (hh << 3);
      float d2 = sqA[M] + sqj - 2.0f * c[v];
      float d  = __builtin_amdgcn_sqrtf(fmaxf(d2, 1e-12f));  // raw v_sqrt_f32
      if (tgA[M] == tj) ap[v] = fmaxf(ap[v], d);
      else              an[v] = fminf(an[v], d);
    }
  }
  // reduce across N (16 lanes within each half-wave)
#pragma unroll
  for (int off = 1; off < 16; off <<= 1) {
#pragma unroll
    for (int v = 0; v < 8; ++v) {
      ap[v] = fmaxf(ap[v], __shfl_xor(ap[v], off, 32));
      an[v] = fminf(an[v], __shfl_xor(an[v], off, 32));
    }
  }
  if (lr == 0) {
#pragma unroll
    for (int v = 0; v < 8; ++v) {
      apL[wave][v + (hh << 3)] = ap[v];
      anL[wave][v + (hh << 3)] = an[v];
    }
  }
  __syncthreads();
  if (tid < 16) {
    float A_ = -INFINITY, N_ = INFINITY;
#pragma unroll
    for (int w = 0; w < 8; ++w) {
      A_ = fmaxf(A_, apL[w][tid]);
      N_ = fminf(N_, anL[w][tid]);
    }
    float hinge = fmaxf(0.0f, MARGIN_V - (N_ - A_));
    atomicAdd(&acc[1], hinge);
  }
}

// ---------------------------------------------------------------------------
// Kernel 3: center loss — only the target column of the distmat is used.
// One wave per row: gather centers[tgt[row]], dot + norm.
// ---------------------------------------------------------------------------
__global__ __launch_bounds__(256) void center_kernel(const float* __restrict__ E,
                                                     const float* __restrict__ C,
                                                     const int* __restrict__ tgt,
                                                     const float* __restrict__ sq,
                                                     float* __restrict__ acc) {
  const int wave = threadIdx.x >> 5;
  const int lane = threadIdx.x & 31;
  const int row  = blockIdx.x * 8 + wave;
  const int t    = tgt[row];
  const float* __restrict__ x = E + (size_t)row * FEAT_DIM;
  const float* __restrict__ c = C + (size_t)t * FEAT_DIM;
  float s1 = 0.0f, s2 = 0.0f;
#pragma unroll
  for (int k = lane; k < FEAT_DIM; k += 32) {
    float xv = x[k], cv = c[k];
    s1 += xv * cv;
    s2 += cv * cv;
  }
#pragma unroll
  for (int off = 16; off > 0; off >>= 1) {
    s1 += __shfl_xor(s1, off, 32);
    s2 += __shfl_xor(s2, off, 32);
  }
  if (lane == 0) {
    float d = sq[row] + s2 - 2.0f * s1;
    d = fminf(fmaxf(d, 1e-12f), 1e12f);
    atomicAdd(&acc[2], d);
  }
}

// ---------------------------------------------------------------------------
// Kernel 4: combine into the scalar output
// ---------------------------------------------------------------------------
__global__ void finalize_kernel(const float* __restrict__ acc, float* __restrict__ out) {
  if (threadIdx.x == 0)
    out[0] = (acc[0] + acc[1] + CENTER_W * acc[2]) * (1.0f / BATCH_N);
}

// ---------------------------------------------------------------------------
extern "C" void kernel_launch(void* const* d_in, const int* in_sizes, int n_in,
                              void* d_out, int out_size, void* d_ws, size_t ws_size,
                              hipStream_t stream) {
  const float* emb     = (const float*)d_in[0];  // [2048, 256]
  const float* outputs = (const float*)d_in[1];  // [2048, 32768]
  const float* centers = (const float*)d_in[2];  // [32768, 256]
  const int*   targets = (const int*)d_in[3];    // [2048]

  float* acc = (float*)d_ws;      // acc[0]=smooth sum, acc[1]=triplet sum, acc[2]=center sum
  float* sq  = acc + 8;           // sq[2048] = ||e_i||^2

  prep_kernel   <<<BATCH_N / 8, 256, 0, stream>>>(emb, sq, acc);
  smooth_kernel <<<BATCH_N,     256, 0, stream>>>(outputs, targets, acc);
  triplet_kernel<<<BATCH_N / 16,256, 0, stream>>>(emb, targets, sq, acc);
  center_kernel <<<BATCH_N / 8, 256, 0, stream>>>(emb, centers, targets, sq, acc);
  finalize_kernel<<<1, 32, 0, stream>>>(acc, (float*)d_out);
}